// MyModel_61933428412966
// MI455X (gfx1250) — compile-verified
//
#include <hip/hip_runtime.h>
#include <hip/hip_bf16.h>

// Reference: BATCH=8, SEQ_LEN=4096, INPUT_SIZE=1024, HIDDEN=4096
//   out[b,h] = h_init[0,h] multiplied by foo, sequentially, SEQ_LEN times.
//   x is never read by the scan body -> skip its 128 MB entirely.
//   Rows are identical across b -> compute HIDDEN chains once, store B copies.

#define RNN_BATCH   8
#define RNN_SEQ_LEN 4096
#define RNN_HIDDEN  4096

__global__ __launch_bounds__(256) void scan_scale_kernel(
    const float* __restrict__ foo,     // d_in[1], 1 element
    const float* __restrict__ h_init,  // d_in[2], HIDDEN elements
    float* __restrict__ out)           // d_out,   BATCH*HIDDEN elements
{
    const int h = blockIdx.x * blockDim.x + threadIdx.x;  // 0..HIDDEN-1
    if (h >= RNN_HIDDEN) return;

    const float f = foo[0];          // uniform scalar, loaded once
    float acc = h_init[h];           // exact starting value per column

    // Exact sequential multiply chain, same rounding order as the reference
    // scan: ((h*f)*f)*...*f. Unroll to cut loop overhead; the dependency
    // chain itself is inherently serial (4096 x v_mul_f32, ~4 cyc each),
    // hidden across the 8 wave32 waves per block / many SIMDs.
#pragma unroll 16
    for (int t = 0; t < RNN_SEQ_LEN; ++t) {
        acc *= f;
    }

    // Broadcast the per-column result to all batch rows. Stores are fully
    // coalesced across threads for each b (stride-1 in h).
#pragma unroll
    for (int b = 0; b < RNN_BATCH; ++b) {
        out[(size_t)b * RNN_HIDDEN + h] = acc;
    }
}

extern "C" void kernel_launch(void* const* d_in, const int* in_sizes, int n_in,
                              void* d_out, int out_size, void* d_ws, size_t ws_size,
                              hipStream_t stream) {
    // Input order per setup_inputs(): x (unused), foo, h_init.
    const float* foo    = (const float*)d_in[1];
    const float* h_init = (const float*)d_in[2];
    float*       out    = (float*)d_out;

    (void)in_sizes; (void)n_in; (void)out_size; (void)d_ws; (void)ws_size;

    const int threads = 256;                          // 8 wave32 waves / block
    const int blocks  = (RNN_HIDDEN + threads - 1) / threads;  // 16 blocks
    scan_scale_kernel<<<blocks, threads, 0, stream>>>(foo, h_init, out);
}